// MultiHeadDGF_38087769981229
// MI455X (gfx1250) — compile-verified
//
#include <hip/hip_runtime.h>
#include <hip/hip_bf16.h>

#define BB  4
#define NN  512
#define DD  128
#define HH  4
#define HID 32
#define EPSF 1e-6f

typedef float v2f __attribute__((ext_vector_type(2)));
typedef float v8f __attribute__((ext_vector_type(8)));

union V8 { v8f v; float f[8]; };

__device__ __forceinline__ v8f wmma_f32(v2f a, v2f b, v8f c) {
    // V_WMMA_F32_16X16X4_F32: D(16x16) = A(16x4) * B(4x16) + C
    return __builtin_amdgcn_wmma_f32_16x16x4_f32(false, a, false, b, (short)0, c,
                                                 false, false);
}

// CDNA5 hardware transcendental tanh (V_TANH_F32) when available
__device__ __forceinline__ float fast_tanh(float x) {
#if __has_builtin(__builtin_amdgcn_tanhf)
    return __builtin_amdgcn_tanhf(x);
#else
    return tanhf(x);
#endif
}

__device__ __forceinline__ float fast_rcp(float x) {
#if __has_builtin(__builtin_amdgcn_rcpf)
    return __builtin_amdgcn_rcpf(x);
#else
    return 1.f / x;
#endif
}

// ---------------------------------------------------------------------------
// Kernel 1: AI[b][h][n][j] = (x @ W1a[h])[n][j] + b1[h][j]
//           AJ[b][h][n][j] = (x @ W1b[h])[n][j]
// one wave per (b,h,16-row tile); 4 WMMA accumulators (2 outputs x 2 col tiles)
// ---------------------------------------------------------------------------
__global__ __launch_bounds__(32)
void k_aiaj(const float* __restrict__ x, const float* __restrict__ W1,
            const float* __restrict__ b1, float* __restrict__ AI,
            float* __restrict__ AJ) {
    int bid  = blockIdx.x;           // BB*HH*(NN/16)
    int nt   = bid & 31;             // NN/16 == 32
    int h    = (bid >> 5) & (HH - 1);
    int b    = bid >> 7;
    int l    = threadIdx.x;
    int rowA = l & 15;
    int colN = l & 15;
    int kOff = (l < 16) ? 0 : 2;

    const float* xrow = x + ((size_t)b * NN + nt * 16 + rowA) * DD;
    const float* W1h  = W1 + (size_t)h * (2 * DD) * HID;

    v8f c0{}, c1{}, c2{}, c3{};
    #pragma unroll 4
    for (int k0 = 0; k0 < DD; k0 += 4) {
        int kk = k0 + kOff;
        v2f a = *(const v2f*)(xrow + kk);                // A: x[n][kk], x[n][kk+1]
        const float* Ba = W1h + (size_t)kk * HID;        // W1a rows kk, kk+1
        const float* Bb = W1h + (size_t)(DD + kk) * HID; // W1b rows kk, kk+1
        v2f b00 = { Ba[colN],      Ba[HID + colN] };
        v2f b01 = { Ba[16 + colN], Ba[HID + 16 + colN] };
        v2f b10 = { Bb[colN],      Bb[HID + colN] };
        v2f b11 = { Bb[16 + colN], Bb[HID + 16 + colN] };
        c0 = wmma_f32(a, b00, c0);
        c1 = wmma_f32(a, b01, c1);
        c2 = wmma_f32(a, b10, c2);
        c3 = wmma_f32(a, b11, c3);
    }

    V8 u0, u1, u2, u3;
    u0.v = c0; u1.v = c1; u2.v = c2; u3.v = c3;
    float* aiOut = AI + ((size_t)(b * HH + h) * NN + nt * 16) * HID;
    float* ajOut = AJ + ((size_t)(b * HH + h) * NN + nt * 16) * HID;
    const float* b1h = b1 + h * HID;
    float bia = b1h[colN], bib = b1h[16 + colN];
    int mBase = (l < 16) ? 0 : 8;
    #pragma unroll
    for (int r = 0; r < 8; ++r) {
        int row = r + mBase;
        aiOut[row * HID + colN]      = u0.f[r] + bia;   // b1 folded into AI
        aiOut[row * HID + 16 + colN] = u1.f[r] + bib;
        ajOut[row * HID + colN]      = u2.f[r];
        ajOut[row * HID + 16 + colN] = u3.f[r];
    }
}

// ---------------------------------------------------------------------------
// Kernel 2: one wave per 16x16 (n,m) tile:
//   gram via f32 WMMA from padded LDS, dist, then per-head
//   sigma = softplus(sum_k tanh(ai+aj)*W2 + b2), adj += exp(-dist/(2s^2+eps))
// ---------------------------------------------------------------------------
__global__ __launch_bounds__(32)
void k_adj(const float* __restrict__ x, const float* __restrict__ AI,
           const float* __restrict__ AJ, const float* __restrict__ W2,
           const float* __restrict__ b2, float* __restrict__ ADJ) {
    __shared__ float xn[16][132];    // pad 128->132: conflict-free WMMA fetch
    __shared__ float xm[16][132];
    __shared__ float x2s[32];
    __shared__ float aih[16][33];    // pad 32->33
    __shared__ float ajh[16][33];
    __shared__ float w2s[HID];

    int b = blockIdx.z, ntile = blockIdx.y, mtile = blockIdx.x;
    int l = threadIdx.x;
    const float* xb = x + (size_t)b * NN * DD;

    // stage x tiles (float4, 16B aligned: 132 % 4 == 0)
    for (int i = l; i < 16 * 32; i += 32) {
        int r = i >> 5, c4 = (i & 31) * 4;
        *(float4*)&xn[r][c4] = *(const float4*)(xb + (size_t)(ntile * 16 + r) * DD + c4);
        *(float4*)&xm[r][c4] = *(const float4*)(xb + (size_t)(mtile * 16 + r) * DD + c4);
    }
    __syncthreads();

    // x2: lanes 0..15 -> n rows, lanes 16..31 -> m rows
    {
        const float* row = (l < 16) ? &xn[l][0] : &xm[l - 16][0];
        float s = 0.f;
        #pragma unroll 4
        for (int k = 0; k < DD; ++k) s += row[k] * row[k];
        x2s[l] = s;
    }
    __syncthreads();

    int rowA = l & 15, colN = l & 15;
    int kOff = (l < 16) ? 0 : 2;

    // gram tile: D(M=n_local, N=m_local) = sum_k xn[M][k] * xm[N][k]
    v8f g{};
    #pragma unroll 4
    for (int k0 = 0; k0 < DD; k0 += 4) {
        int kk = k0 + kOff;
        v2f a  = *(const v2f*)&xn[rowA][kk];
        v2f bv = *(const v2f*)&xm[colN][kk];
        g = wmma_f32(a, bv, g);
    }
    V8 gu; gu.v = g;

    int mBase = (l < 16) ? 0 : 8;
    float dist[8], acc[8];
    #pragma unroll
    for (int r = 0; r < 8; ++r) {
        int nl = r + mBase;
        float d = x2s[nl] + x2s[16 + colN] - 2.f * gu.f[r];
        dist[r] = fmaxf(d, 0.f);
        acc[r]  = 0.f;
    }

    for (int h = 0; h < HH; ++h) {
        __syncthreads();   // previous-iteration readers done before overwrite
        const float* aiSrc = AI + ((size_t)(b * HH + h) * NN + ntile * 16) * HID;
        const float* ajSrc = AJ + ((size_t)(b * HH + h) * NN + mtile * 16) * HID;
        for (int i = l; i < 16 * HID; i += 32) {
            int r = i >> 5, c = i & 31;
            aih[r][c] = aiSrc[r * HID + c];
            ajh[r][c] = ajSrc[r * HID + c];
        }
        w2s[l] = W2[h * HID + l];
        __syncthreads();
        float b2h = b2[h];

        #pragma unroll
        for (int r = 0; r < 8; ++r) {
            int nl = r + mBase;
            float z = 0.f;
            #pragma unroll 8
            for (int k = 0; k < HID; ++k)
                z += fast_tanh(aih[nl][k] + ajh[colN][k]) * w2s[k];
            z += b2h;
            // softplus via hardware exp/log (z is O(0.1) here)
            float sp = (z > 15.f) ? z : __logf(1.f + __expf(z));
            float denom = 2.f * sp * sp + EPSF;
            acc[r] += __expf(-dist[r] * fast_rcp(denom));
        }
    }

    float* out = ADJ + ((size_t)b * NN + ntile * 16) * NN + mtile * 16;
    #pragma unroll
    for (int r = 0; r < 8; ++r)
        out[(size_t)(r + mBase) * NN + colN] = acc[r] * (1.f / HH);
}

// ---------------------------------------------------------------------------
// Kernel 3: HOUT = ADJ @ x   (K = 512, operands streamed from L2)
// ---------------------------------------------------------------------------
__global__ __launch_bounds__(32)
void k_hout(const float* __restrict__ ADJ, const float* __restrict__ x,
            float* __restrict__ HOUT) {
    int b = blockIdx.z, nt = blockIdx.y, dt = blockIdx.x;  // dt < DD/16
    int l = threadIdx.x;
    int rowA = l & 15, colN = l & 15;
    int kOff = (l < 16) ? 0 : 2;

    const float* arow = ADJ + ((size_t)b * NN + nt * 16 + rowA) * NN;
    const float* xcol = x + (size_t)b * NN * DD + dt * 16 + colN;

    v8f c{};
    #pragma unroll 4
    for (int k0 = 0; k0 < NN; k0 += 4) {
        int kk = k0 + kOff;
        v2f a  = *(const v2f*)(arow + kk);                       // adj[n][kk..kk+1]
        v2f bv = { xcol[(size_t)kk * DD], xcol[(size_t)(kk + 1) * DD] };
        c = wmma_f32(a, bv, c);
    }
    V8 u; u.v = c;
    int mBase = (l < 16) ? 0 : 8;
    float* out = HOUT + ((size_t)b * NN + nt * 16) * DD + dt * 16;
    #pragma unroll
    for (int r = 0; r < 8; ++r)
        out[(size_t)(r + mBase) * DD + colN] = u.f[r];
}

// ---------------------------------------------------------------------------
// Kernel 4: out = HOUT @ Wp + bp  (bias preloaded into the WMMA accumulator)
// ---------------------------------------------------------------------------
__global__ __launch_bounds__(32)
void k_proj(const float* __restrict__ HOUT, const float* __restrict__ Wp,
            const float* __restrict__ bp, float* __restrict__ out) {
    int b = blockIdx.z, nt = blockIdx.y, dt = blockIdx.x;
    int l = threadIdx.x;
    int rowA = l & 15, colN = l & 15;
    int kOff = (l < 16) ? 0 : 2;

    const float* hrow = HOUT + ((size_t)b * NN + nt * 16 + rowA) * DD;
    const float* wcol = Wp + dt * 16 + colN;
    float bpv = bp[dt * 16 + colN];                // C element's N-index == out col
    v8f c = { bpv, bpv, bpv, bpv, bpv, bpv, bpv, bpv };

    #pragma unroll 4
    for (int k0 = 0; k0 < DD; k0 += 4) {
        int kk = k0 + kOff;
        v2f a  = *(const v2f*)(hrow + kk);
        v2f bv = { wcol[(size_t)kk * DD], wcol[(size_t)(kk + 1) * DD] };
        c = wmma_f32(a, bv, c);
    }
    V8 u; u.v = c;
    int mBase = (l < 16) ? 0 : 8;
    float* o = out + ((size_t)b * NN + nt * 16) * DD + dt * 16;
    #pragma unroll
    for (int r = 0; r < 8; ++r)
        o[(size_t)(r + mBase) * DD + colN] = u.f[r];
}

// ---------------------------------------------------------------------------
extern "C" void kernel_launch(void* const* d_in, const int* in_sizes, int n_in,
                              void* d_out, int out_size, void* d_ws, size_t ws_size,
                              hipStream_t stream) {
    const float* x  = (const float*)d_in[0];
    const float* W1 = (const float*)d_in[1];
    const float* b1 = (const float*)d_in[2];
    const float* W2 = (const float*)d_in[3];
    const float* b2 = (const float*)d_in[4];
    const float* Wp = (const float*)d_in[5];
    const float* bp = (const float*)d_in[6];
    float* out = (float*)d_out;

    // workspace layout (floats): AI | AJ | ADJ | HOUT  == 7 MiB total
    float* AI   = (float*)d_ws;
    float* AJ   = AI  + (size_t)BB * HH * NN * HID;
    float* ADJ  = AJ  + (size_t)BB * HH * NN * HID;
    float* HOUT = ADJ + (size_t)BB * NN * NN;

    k_aiaj<<<BB * HH * (NN / 16), 32, 0, stream>>>(x, W1, b1, AI, AJ);
    k_adj <<<dim3(NN / 16, NN / 16, BB), 32, 0, stream>>>(x, AI, AJ, W2, b2, ADJ);
    k_hout<<<dim3(DD / 16, NN / 16, BB), 32, 0, stream>>>(ADJ, x, HOUT);
    k_proj<<<dim3(DD / 16, NN / 16, BB), 32, 0, stream>>>(HOUT, Wp, bp, out);
}